// LorentzBatchNorm2d_3719441678457
// MI455X (gfx1250) — compile-verified
//
#include <hip/hip_runtime.h>
#include <math.h>

// LorentzBatchNorm2d for x:(64,64,64,64) f32.
// Bandwidth-bound: ~134 MB HBM (x read once, L2-resident after; out written once).
// WMMA f32 16x16x4 used for the per-row Lorentz inner products (full f32 precision).

#define Bn 64
#define Cn 64
#define HW 4096              // H*W
#define NROW (Bn * HW)       // 262144 rows
#define EPSf 1e-5f

// d_ws float offsets
#define WS_MU     0          // 64 floats: mu
#define WS_SCALE  64         // 1 float : 1/(sqrt(var)+eps)
#define WS_MPART  128        // 4096 floats: per-(b,c)-plane partial sums
#define WS_VPART  4352       // 2048 floats: per-block partial sums of d^2

typedef float v2f __attribute__((ext_vector_type(2)));
typedef float v8f __attribute__((ext_vector_type(8)));

// ---- Pass 1: per-(b,c) plane sums (deterministic two-stage mean) ----
__global__ void k1_plane_sums(const float* __restrict__ x, float* __restrict__ ws) {
    __shared__ float red[256];
    const int p = blockIdx.x;                    // plane index = b*64 + c
    const float* base = x + (size_t)p * HW;
    float s = 0.0f;
    for (int i = threadIdx.x; i < HW; i += 256) {
        __builtin_prefetch(base + i + 2048, 0, 1);   // global_prefetch_b8 (L2 warm)
        s += base[i];
    }
    red[threadIdx.x] = s;
    __syncthreads();
    for (int off = 128; off > 0; off >>= 1) {
        if (threadIdx.x < off) red[threadIdx.x] += red[threadIdx.x + off];
        __syncthreads();
    }
    if (threadIdx.x == 0) ws[WS_MPART + p] = red[0];
}

// ---- Pass 2: reduce plane sums -> mu (single block of 64) ----
__global__ void k2_mu(float* __restrict__ ws) {
    __shared__ float lm[64];
    __shared__ float rs;
    const int t = threadIdx.x;
    if (t < 64) {
        float s = 0.0f;
        for (int b = 0; b < Bn; ++b) s += ws[WS_MPART + b * 64 + t];
        lm[t] = s / (float)NROW;                 // m[c]
    }
    __syncthreads();
    if (t == 0) {
        float mm = lm[0] * lm[0];
        for (int c = 1; c < 64; ++c) mm -= lm[c] * lm[c];
        rs = 1.0f / sqrtf(fmaxf(mm, EPSf));
    }
    __syncthreads();
    if (t < 64) ws[WS_MU + t] = lm[t] * rs;      // mu[c]
}

// ---- Pass 3: alpha via WMMA (f32 16x16x4), accumulate d^2 partials ----
// A = signed-mu replicated over all 16 rows (documented 16x4 f32 A layout):
//   lanes 0-15 hold K=k0,k0+1 ; lanes 16-31 hold K=k0+2,k0+3.
// B = 16 rows of x (4x16), K split mirrors A, N = lane%16.
// All A rows identical => every D column n equals alpha[n0+n]; lane l reads
// alpha for row n0+(l&15) from acc[0].
__global__ void k3_var(const float* __restrict__ x, float* __restrict__ ws) {
    __shared__ float mu_s[64];
    __shared__ float red[256];
    const int tid = threadIdx.x;
    if (tid < 64) {
        float m = ws[WS_MU + tid];
        mu_s[tid] = (tid == 0) ? m : -m;         // Lorentz signature
    }
    __syncthreads();

    const int wave = tid >> 5;
    const int lane = tid & 31;
    const int half = lane >> 4;                  // 0: K=k0,k0+1 ; 1: K=k0+2,k0+3
    const int nl   = lane & 15;                  // N index (row within tile)
    const int n0   = (blockIdx.x * 8 + wave) * 16;
    const int bb   = n0 >> 12;                   // / 4096 (tile never crosses b)
    const int hw0  = n0 & 4095;
    const float* base = x + (size_t)bb * (Cn * HW) + hw0;

    v8f acc = {0.f, 0.f, 0.f, 0.f, 0.f, 0.f, 0.f, 0.f};
#pragma unroll
    for (int k0 = 0; k0 < 64; k0 += 4) {
        v2f a, b;
        a.x = mu_s[k0 + 2 * half];
        a.y = mu_s[k0 + 2 * half + 1];
        b.x = base[(size_t)(k0 + 2 * half)     * HW + nl];   // coalesced 64B/half-wave
        b.y = base[(size_t)(k0 + 2 * half + 1) * HW + nl];
        acc = __builtin_amdgcn_wmma_f32_16x16x4_f32(
                  false, a, false, b, (short)0, acc, false, false);
    }

    float alpha = fmaxf(acc[0], 1.0f + EPSf);
    float r = sqrtf(alpha * alpha - 1.0f);
    float d = logf(alpha + r);                   // acosh(alpha)
    red[tid] = (half == 0) ? d * d : 0.0f;       // hi half duplicates lo half
    __syncthreads();
    for (int off = 128; off > 0; off >>= 1) {
        if (tid < off) red[tid] += red[tid + off];
        __syncthreads();
    }
    if (tid == 0) ws[WS_VPART + blockIdx.x] = red[0];
}

// ---- Pass 4: reduce d^2 partials -> scale ----
__global__ void k4_scale(float* __restrict__ ws) {
    __shared__ float red[256];
    const int t = threadIdx.x;
    float s = 0.0f;
    for (int i = t; i < 2048; i += 256) s += ws[WS_VPART + i];
    red[t] = s;
    __syncthreads();
    for (int off = 128; off > 0; off >>= 1) {
        if (t < off) red[t] += red[t + off];
        __syncthreads();
    }
    if (t == 0) {
        float var = red[0] / (float)NROW;
        ws[WS_SCALE] = 1.0f / (sqrtf(var) + EPSf);
    }
}

// ---- Pass 5: output. One lane per row; x row held in 64 VGPRs (single read,
// L2-resident), alpha recomputed in-register; all global I/O coalesced. ----
__global__ void k5_out(const float* __restrict__ x,
                       const float* __restrict__ gamma,
                       const float* __restrict__ beta,
                       const float* __restrict__ ws,
                       float* __restrict__ out) {
    __shared__ float mu[64];
    __shared__ float gg[63];
    __shared__ float bt[63];
    __shared__ float sc;
    const int tid = threadIdx.x;
    if (tid < 64) mu[tid] = ws[WS_MU + tid];
    if (tid < 63) { gg[tid] = gamma[tid]; bt[tid] = beta[tid]; }
    if (tid == 0) sc = ws[WS_SCALE];
    __syncthreads();

    const int n  = blockIdx.x * 256 + tid;
    const int bb = n >> 12;
    const int hw = n & 4095;
    const size_t base = (size_t)bb * (Cn * HW) + hw;

    float xr[64];
#pragma unroll
    for (int c = 0; c < 64; ++c) xr[c] = x[base + (size_t)c * HW];

    float alpha = xr[0] * mu[0];
#pragma unroll
    for (int c = 1; c < 64; ++c) alpha -= xr[c] * mu[c];
    alpha = fmaxf(alpha, 1.0f + EPSf);

    const float r    = sqrtf(alpha * alpha - 1.0f);
    const float d    = logf(alpha + r);          // acosh
    const float coef = d / r;
    const float v0   = coef * (xr[0] - alpha * mu[0]);
    const float f    = v0 / (1.0f + mu[0]);      // vo[:,0] == 0 identically

    float acc2 = 0.0f;
#pragma unroll
    for (int c = 1; c < 64; ++c) {
        float v  = coef * (xr[c] - alpha * mu[c]);
        float vo = v - f * mu[c];
        float s  = gg[c - 1] * vo * sc + bt[c - 1];
        acc2 += s * s;
        out[base + (size_t)c * HW] = s;
    }
    out[base] = sqrtf(1.0f + acc2);
}

extern "C" void kernel_launch(void* const* d_in, const int* in_sizes, int n_in,
                              void* d_out, int out_size, void* d_ws, size_t ws_size,
                              hipStream_t stream) {
    const float* x     = (const float*)d_in[0];
    const float* gamma = (const float*)d_in[1];
    const float* beta  = (const float*)d_in[2];
    float* out = (float*)d_out;
    float* ws  = (float*)d_ws;                   // needs 25.6 KB

    k1_plane_sums<<<Bn * Cn, 256, 0, stream>>>(x, ws);          // 4096 blocks
    k2_mu        <<<1, 64, 0, stream>>>(ws);
    k3_var       <<<NROW / 128, 256, 0, stream>>>(x, ws);       // 2048 blocks
    k4_scale     <<<1, 256, 0, stream>>>(ws);
    k5_out       <<<NROW / 256, 256, 0, stream>>>(x, gamma, beta, ws, out); // 1024 blocks
}